// AlgelogicNetwork_12455405158468
// MI455X (gfx1250) — compile-verified
//
#include <hip/hip_runtime.h>
#include <hip/hip_bf16.h>
#include <math.h>

// Problem constants from the reference
#define M_  16
#define J_  2
#define I_  3
#define L_  2
#define W_  9
#define STEEP 10.0f

typedef __attribute__((ext_vector_type(2))) float v2f;
typedef __attribute__((ext_vector_type(8))) float v8f;

// One wave32 does everything:
//   lanes 0..31  <-> (m,j) pairs, m = lane>>1, j = lane&1
// match_q computed via 10 rank-1 V_WMMA_F32_16X16X4_F32 accumulations
// (only K=0 active in A and B -> immune to B-matrix K-slot layout ambiguity).
// All WMMA operands are pre-loaded into registers unconditionally and masked
// with a {0,1} multiply so no EXEC-predicated loads land inside the WMMA chain.
__global__ __launch_bounds__(32)
void algelogic_kernel(const float* __restrict__ state,      // (1, W*L) = 18
                      const float* __restrict__ constants,  // (M, J+1, L) = 96
                      const float* __restrict__ gammas,     // (M, J+1, L) = 96
                      const float* __restrict__ head_w,     // (M, J, I, L) = 192
                      const float* __restrict__ tail_w,     // (M, L, I) = 96
                      const float* __restrict__ tail_b,     // (M, L) = 32
                      float* __restrict__ out)              // (M,) = 16
{
    __shared__ float s_wm[16][L_];      // wm padded to 16 rows (rows 9..15 = 0)
    __shared__ float s_coef[32][5];     // per-(m,j) rank-term A coefficients
    __shared__ float s_matq[32][16];    // match_q, padded to 32x16
    __shared__ float s_picked[32][I_];  // picked captures per (m,j)
    __shared__ float s_P[M_];           // per-m norm

    const int lane = threadIdx.x;
    const int m = lane >> 1;
    const int j = lane & 1;

    // ---- stage wm into LDS (zero-pad rows 9..15) ------------------------
    if (lane < 16) { s_wm[lane][0] = 0.0f; s_wm[lane][1] = 0.0f; }
    // single wave: program order guarantees the zeroing stores precede these
    if (lane < W_ * L_) s_wm[lane >> 1][lane & 1] = state[lane];

    // ---- per-(m,j) gate coefficients ------------------------------------
    // g = gammas[m, j+1, :]; c = constants[m, j, :]
    const float g0 = gammas[m * 6 + (j + 1) * 2 + 0];
    const float g1 = gammas[m * 6 + (j + 1) * 2 + 1];
    const float c0 = constants[m * 6 + j * 2 + 0];
    const float c1 = constants[m * 6 + j * 2 + 1];
    const float sig0 = 1.0f / (1.0f + expf(-STEEP * (g0 - 0.5f)));
    const float sig1 = 1.0f / (1.0f + expf(-STEEP * (g1 - 0.5f)));
    const float mask0 = (g0 > 0.5f) ? 1.0f : 0.0f;
    const float mask1 = (g1 > 0.5f) ? 1.0f : 0.0f;

    // match_q[mj,w] = sig0*(c0-wm0)^2 + sig1*(c1-wm1)^2
    //   = sig0*wm0^2 + sig1*wm1^2 - 2sig0c0*wm0 - 2sig1c1*wm1 + (sig0c0^2+sig1c1^2)
    s_coef[lane][0] = sig0;
    s_coef[lane][1] = sig1;
    s_coef[lane][2] = -2.0f * sig0 * c0;
    s_coef[lane][3] = -2.0f * sig1 * c1;
    s_coef[lane][4] = sig0 * c0 * c0 + sig1 * c1 * c1;

    __syncthreads();

    // ---- gather all WMMA operand data unconditionally (no EXEC games) ---
    const int   n   = lane & 15;                   // valid index for ALL lanes
    const float sel = (lane < 16) ? 1.0f : 0.0f;   // mask applied via VALU mul
    const float w0 = s_wm[n][0];
    const float w1 = s_wm[n][1];

    float ca[5], cb[5], bv[5];
#pragma unroll
    for (int t = 0; t < 5; ++t) {
        ca[t] = s_coef[n][t];        // rows mj = 0..15   (contig -> b128+b32)
        cb[t] = s_coef[16 + n][t];   // rows mj = 16..31
    }
    bv[0] = w0 * w0;
    bv[1] = w1 * w1;
    bv[2] = w0;
    bv[3] = w1;
    bv[4] = (n < W_) ? 1.0f : 0.0f;  // pad columns -> match_q = 0 (never read)

    // ---- match_q via rank-1 WMMA accumulation ---------------------------
    // A (16x4 f32): VGPR0 lanes0-15 = (M=lane, K=0)  [documented]; all other
    // K slots forced to 0.  B (4x16 f32): (K=0, N=lane) at VGPR0 lanes0-15;
    // all other slots 0 -> K-slot layout ambiguity is irrelevant.
    v8f acc0 = {0.f, 0.f, 0.f, 0.f, 0.f, 0.f, 0.f, 0.f}; // rows mj = 0..15
    v8f acc1 = {0.f, 0.f, 0.f, 0.f, 0.f, 0.f, 0.f, 0.f}; // rows mj = 16..31

#pragma unroll
    for (int t = 0; t < 5; ++t) {
        v2f a0, a1, b;
        a0.x = ca[t] * sel;  a0.y = 0.0f;
        a1.x = cb[t] * sel;  a1.y = 0.0f;
        b.x  = bv[t] * sel;  b.y  = 0.0f;
        acc0 = __builtin_amdgcn_wmma_f32_16x16x4_f32(
                   false, a0, false, b, (short)0, acc0, false, false);
        acc1 = __builtin_amdgcn_wmma_f32_16x16x4_f32(
                   false, a1, false, b, (short)0, acc1, false, false);
    }

    // D layout: VGPR v -> row (v + (lane<16 ? 0 : 8)), col = lane&15
    {
        const int rbase = (lane < 16) ? 0 : 8;
#pragma unroll
        for (int v = 0; v < 8; ++v) {
            s_matq[rbase + v][n]      = acc0[v];
            s_matq[16 + rbase + v][n] = acc1[v];
        }
    }
    __syncthreads();

    // ---- argmin over w (first-index tie-break, like jnp.argmin) ---------
    int best = 0;
    {
        float bq = s_matq[lane][0];
#pragma unroll
        for (int w = 1; w < W_; ++w) {
            const float q = s_matq[lane][w];
            if (q < bq) { bq = q; best = w; }
        }
    }

    // ---- picked captures: mask * head_w . wm[best] ----------------------
    {
        const float wb0 = s_wm[best][0];
        const float wb1 = s_wm[best][1];
        const int hb = m * (J_ * I_ * L_) + j * (I_ * L_);
#pragma unroll
        for (int i = 0; i < I_; ++i) {
            s_picked[lane][i] = mask0 * head_w[hb + i * L_ + 0] * wb0
                              + mask1 * head_w[hb + i * L_ + 1] * wb1;
        }
    }
    __syncthreads();

    // ---- tail GEMV + norm (lanes 0..15, one lane per m) -----------------
    if (lane < M_) {
        float cap[I_];
#pragma unroll
        for (int i = 0; i < I_; ++i)
            cap[i] = s_picked[2 * lane][i] + s_picked[2 * lane + 1][i];

        float nrm2 = 0.0f;
#pragma unroll
        for (int l = 0; l < L_; ++l) {
            float acc = tail_b[lane * L_ + l];
#pragma unroll
            for (int i = 0; i < I_; ++i)
                acc += tail_w[lane * (L_ * I_) + l * I_ + i] * cap[i];
            nrm2 += acc * acc;
        }
        s_P[lane] = sqrtf(nrm2);
    }
    __syncthreads();

    // ---- softmax over the 16 P values -----------------------------------
    if (lane < M_) {
        float mx = s_P[0];
#pragma unroll
        for (int k = 1; k < M_; ++k) mx = fmaxf(mx, s_P[k]);
        float sum = 0.0f;
#pragma unroll
        for (int k = 0; k < M_; ++k) sum += expf(s_P[k] - mx);
        out[lane] = expf(s_P[lane] - mx) / sum;
    }
}

extern "C" void kernel_launch(void* const* d_in, const int* in_sizes, int n_in,
                              void* d_out, int out_size, void* d_ws, size_t ws_size,
                              hipStream_t stream) {
    const float* state     = (const float*)d_in[0];
    const float* constants = (const float*)d_in[1];
    const float* gammas    = (const float*)d_in[2];
    const float* head_w    = (const float*)d_in[3];
    const float* tail_w    = (const float*)d_in[4];
    const float* tail_b    = (const float*)d_in[5];
    float* out = (float*)d_out;
    (void)in_sizes; (void)n_in; (void)out_size; (void)d_ws; (void)ws_size;

    algelogic_kernel<<<dim3(1), dim3(32), 0, stream>>>(
        state, constants, gammas, head_w, tail_w, tail_b, out);
}